// MambaLayer_12068858101830
// MI455X (gfx1250) — compile-verified
//
#include <hip/hip_runtime.h>
#include <hip/hip_bf16.h>

typedef __attribute__((ext_vector_type(2))) float v2f;
typedef __attribute__((ext_vector_type(8))) float v8f;

#define NSEQ   16
#define LSEQ   4096
#define DINNER 192
#define DSTATE 16
#define DTRANK 6
#define CDIM   96
#define NPIX   16384          // B*H*W
#define XDBL_LD 48            // padded 38 -> 48
#define NCHUNK 64
#define TCHUNK 64             // NCHUNK * TCHUNK == LSEQ

// ---------------------------------------------------------------------------
// gfx1250 async copy: global -> LDS, bypassing VGPRs (ASYNCcnt tracked).
// dsaddr = low 32 bits of the generic shared pointer (LDS byte offset).
// ---------------------------------------------------------------------------
__device__ __forceinline__ void async_ld_lds_b32(void* lds_dst, const void* gsrc) {
  unsigned int dsaddr = (unsigned int)(unsigned long long)lds_dst;
  asm volatile("global_load_async_to_lds_b32 %0, %1, off"
               :: "v"(dsaddr), "v"(gsrc)
               : "memory");
}
__device__ __forceinline__ void wait_asynccnt0() {
  asm volatile("s_wait_asynccnt 0" ::: "memory");
}

// ---------------------------------------------------------------------------
// K1: per-pixel LayerNorm + gate MLP (96 -> 24 -> 4 -> softmax)
// one block (96 threads) per pixel
// ---------------------------------------------------------------------------
__global__ void ln_gate_kernel(const float* __restrict__ x,
                               const float* __restrict__ ng, const float* __restrict__ nb,
                               const float* __restrict__ w1, const float* __restrict__ b1,
                               const float* __restrict__ w2, const float* __restrict__ b2,
                               float* __restrict__ xn, float* __restrict__ gate) {
  __shared__ float s_v[CDIM];
  __shared__ float s_x[CDIM];
  __shared__ float s_t[24];
  __shared__ float s_stats[2];
  int pix = blockIdx.x;
  int c   = threadIdx.x;
  int b   = pix >> 12;
  int rem = pix & 4095;
  int hh  = rem >> 6;
  int ww  = rem & 63;
  float v = x[((size_t)(b * CDIM + c) * 64 + hh) * 64 + ww];
  s_v[c] = v;
  __syncthreads();
  if (c == 0) {
    float s = 0.f, s2 = 0.f;
    for (int i = 0; i < CDIM; ++i) { s += s_v[i]; s2 += s_v[i] * s_v[i]; }
    float mu  = s * (1.f / CDIM);
    float var = s2 * (1.f / CDIM) - mu * mu;
    s_stats[0] = mu;
    s_stats[1] = rsqrtf(var + 1e-5f);
  }
  __syncthreads();
  float xv = (v - s_stats[0]) * s_stats[1] * ng[c] + nb[c];
  s_x[c] = xv;
  xn[(size_t)pix * CDIM + c] = xv;
  __syncthreads();
  if (c < 24) {
    float t = b1[c];
    for (int k = 0; k < CDIM; ++k) t = fmaf(w1[c * CDIM + k], s_x[k], t);
    s_t[c] = tanhf(t);
  }
  __syncthreads();
  if (c < 4) {
    float g = b2[c];
    for (int k = 0; k < 24; ++k) g = fmaf(w2[c * 24 + k], s_t[k], g);
    s_v[c] = g; // reuse
  }
  __syncthreads();
  if (c == 0) {
    float m  = fmaxf(fmaxf(s_v[0], s_v[1]), fmaxf(s_v[2], s_v[3]));
    float e0 = __expf(s_v[0] - m), e1 = __expf(s_v[1] - m);
    float e2 = __expf(s_v[2] - m), e3 = __expf(s_v[3] - m);
    float inv = 1.f / (e0 + e1 + e2 + e3);
    gate[(size_t)pix * 4 + 0] = e0 * inv;
    gate[(size_t)pix * 4 + 1] = e1 * inv;
    gate[(size_t)pix * 4 + 2] = e2 * inv;
    gate[(size_t)pix * 4 + 3] = e3 * inv;
  }
}

// ---------------------------------------------------------------------------
// K2: in_proj WMMA GEMM for dirs 0,1 only (dirs 2,3 are reversals).
// xz01[row, 0..383], row = (dir*4+b)*4096 + l, dir in {0,1}.
// One wave per 16x16 tile; K=96 -> 24 x v_wmma_f32_16x16x4_f32.
// ---------------------------------------------------------------------------
__global__ void inproj_wmma(const float* __restrict__ xn, const float* __restrict__ Wp,
                            float* __restrict__ xz01) {
  int lane = threadIdx.x;
  int lm = lane & 15;
  int lk = (lane >> 4) * 2;           // 0 or 2
  int m0 = blockIdx.x << 4;
  int n0 = blockIdx.y << 4;
  int row = m0 + lm;
  int s2  = row >> 12;                // dir*4 + b  (0..7)
  int l   = row & 4095;
  int bb  = s2 & 3;
  int pix = (s2 < 4) ? ((bb << 12) + l)
                     : ((bb << 12) + ((l & 63) << 6) + (l >> 6));
  const float* ar = xn + (size_t)pix * CDIM;
  const float* br = Wp + (size_t)(n0 + lm) * CDIM;
  v8f acc = {};
#pragma unroll
  for (int k0 = 0; k0 < CDIM; k0 += 4) {
    v2f a, b;
    a.x = ar[k0 + lk];
    a.y = ar[k0 + lk + 1];
    b.x = br[k0 + lk];
    b.y = br[k0 + lk + 1];
    acc = __builtin_amdgcn_wmma_f32_16x16x4_f32(false, a, false, b, (short)0, acc,
                                                false, false);
  }
  int rb = m0 + ((lane >> 4) << 3);
#pragma unroll
  for (int r = 0; r < 8; ++r)
    xz01[(size_t)(rb + r) * 384 + n0 + lm] = acc[r];
}

// ---------------------------------------------------------------------------
// Generic WMMA GEMM:  C[M,N] = A[M,K] * Bw[N,K]^T   (weights row-major (N,K))
// One wave per 16x16 tile; Nact masks padded output columns.
// ---------------------------------------------------------------------------
__global__ void gemm_wmma_nt(const float* __restrict__ A, int lda,
                             const float* __restrict__ Bw, int ldb, int Nact,
                             float* __restrict__ C, int ldc, int K) {
  int lane = threadIdx.x;
  int lm = lane & 15;
  int lk = (lane >> 4) * 2;
  int m0 = blockIdx.x << 4;
  int n0 = blockIdx.y << 4;
  const float* ar = A + (size_t)(m0 + lm) * lda;
  int nn = n0 + lm;
  const float* br = Bw + (size_t)nn * ldb;
  bool nok = (nn < Nact);
  v8f acc = {};
  for (int k0 = 0; k0 < K; k0 += 4) {
    v2f a, b;
    a.x = ar[k0 + lk];
    a.y = ar[k0 + lk + 1];
    if (nok) {
      b.x = br[k0 + lk];
      b.y = br[k0 + lk + 1];
    } else {
      b.x = 0.f;
      b.y = 0.f;
    }
    acc = __builtin_amdgcn_wmma_f32_16x16x4_f32(false, a, false, b, (short)0, acc,
                                                false, false);
  }
  int rb = m0 + ((lane >> 4) << 3);
#pragma unroll
  for (int r = 0; r < 8; ++r)
    C[(size_t)(rb + r) * ldc + n0 + lm] = acc[r];
}

// ---------------------------------------------------------------------------
// K3: depthwise causal conv1d (taps=4) + SiLU, all 16 sequences.
// Reads xs from xz01 (cols 0..191) with reversal mapping for dirs 2,3.
// xc layout: (s*4096 + l, 192)
// ---------------------------------------------------------------------------
__global__ void conv_silu_kernel(const float* __restrict__ xz01,
                                 const float* __restrict__ cw, const float* __restrict__ cb,
                                 float* __restrict__ xc) {
  long gid = (long)blockIdx.x * blockDim.x + threadIdx.x;
  if (gid >= (long)NSEQ * LSEQ * DINNER) return;
  int d = (int)(gid % DINNER);
  int l = (int)((gid / DINNER) & 4095);
  int s = (int)(gid / ((long)DINNER * LSEQ));
  int dir = s >> 2, bb = s & 3;
  float acc = cb[d];
#pragma unroll
  for (int j = 0; j < 4; ++j) {
    int lj = l - 3 + j;
    if (lj >= 0) {
      int row = (dir < 2) ? ((dir * 4 + bb) * LSEQ + lj)
                          : (((dir - 2) * 4 + bb) * LSEQ + (4095 - lj));
      acc = fmaf(cw[d * 4 + j], xz01[(size_t)row * 384 + d], acc);
    }
  }
  acc = acc / (1.f + __expf(-acc)); // SiLU
  xc[gid] = acc;
}

// ---------------------------------------------------------------------------
// Chunked selective scan: h_t = a_t*h_{t-1} + b_t is affine with diagonal a_t,
// so split L into 64 chunks of 64, compute per-chunk (P = prod a, H_loc from
// h=0) in parallel, prefix-combine across chunks, then re-run chunks seeded
// with the correct incoming state.
//
// Pass 1: one block per (seq, chunk); 192 threads = one d-channel each.
// x_dbl chunk staged into LDS via gfx1250 async-to-LDS loads.
// ---------------------------------------------------------------------------
__global__ void scan_pass1(const float* __restrict__ xc,
                           const float* __restrict__ xdbl,
                           const float* __restrict__ dtw, const float* __restrict__ dtb,
                           const float* __restrict__ Alog,
                           float* __restrict__ P, float* __restrict__ Hloc) {
  __shared__ float sh[TCHUNK][40]; // [t][0..5]=dt, [6..21]=B, [22..37]=C
  int s   = blockIdx.x;
  int cch = blockIdx.y;
  int d   = threadIdx.x;
  int t0  = cch * TCHUNK;
  for (int i = d; i < TCHUNK * 38; i += DINNER) {
    int t = i / 38, k = i % 38;
    async_ld_lds_b32(&sh[t][k],
                     &xdbl[(size_t)(s * LSEQ + t0 + t) * XDBL_LD + k]);
  }
  wait_asynccnt0();
  __syncthreads();

  float Av[DSTATE], h[DSTATE], p[DSTATE];
#pragma unroll
  for (int j = 0; j < DSTATE; ++j) {
    Av[j] = -__expf(Alog[d * DSTATE + j]);
    h[j]  = 0.f;
    p[j]  = 1.f;
  }
  float wdt[DTRANK];
#pragma unroll
  for (int r = 0; r < DTRANK; ++r) wdt[r] = dtw[d * DTRANK + r];
  float bdt = dtb[d];

  for (int tt = 0; tt < TCHUNK; ++tt) {
    size_t row = (size_t)s * LSEQ + t0 + tt;
    float u = xc[row * DINNER + d];
    float dtv = bdt;
#pragma unroll
    for (int r = 0; r < DTRANK; ++r) dtv = fmaf(wdt[r], sh[tt][r], dtv);
    float delta = (dtv > 20.f) ? dtv : log1pf(__expf(dtv));
    float du = delta * u;
#pragma unroll
    for (int j = 0; j < DSTATE; ++j) {
      float aj = __expf(delta * Av[j]);
      h[j] = fmaf(aj, h[j], du * sh[tt][6 + j]);
      p[j] *= aj;
    }
  }
  size_t base = (((size_t)s * NCHUNK + cch) * DINNER + d) * DSTATE;
#pragma unroll
  for (int j = 0; j < DSTATE; ++j) {
    P[base + j]    = p[j];
    Hloc[base + j] = h[j];
  }
}

// Sequential prefix across chunks, parallel over (s, d, j) = 49152 lanes.
// Overwrites Hloc[c] with the INCOMING state for chunk c.
__global__ void scan_combine(const float* __restrict__ P, float* __restrict__ Hloc) {
  int gid = blockIdx.x * blockDim.x + threadIdx.x;
  if (gid >= NSEQ * DINNER * DSTATE) return;
  int j = gid & (DSTATE - 1);
  int d = (gid >> 4) % DINNER;
  int s = gid / (DINNER * DSTATE);
  float h = 0.f;
  for (int c = 0; c < NCHUNK; ++c) {
    size_t idx = (((size_t)s * NCHUNK + c) * DINNER + d) * DSTATE + j;
    float p  = P[idx];
    float hl = Hloc[idx];
    Hloc[idx] = h;            // h_in for chunk c
    h = fmaf(p, h, hl);
  }
}

// Pass 2: re-run each chunk with the correct incoming state; emit
// y = (scan_y + D*u) * silu(z) in place over xc.
__global__ void scan_pass2(float* __restrict__ xc,
                           const float* __restrict__ xdbl,
                           const float* __restrict__ xz01,
                           const float* __restrict__ dtw, const float* __restrict__ dtb,
                           const float* __restrict__ Alog, const float* __restrict__ Dp,
                           const float* __restrict__ Hin) {
  __shared__ float sh[TCHUNK][40];
  int s   = blockIdx.x;
  int cch = blockIdx.y;
  int d   = threadIdx.x;
  int t0  = cch * TCHUNK;
  int dir = s >> 2, bb = s & 3;
  for (int i = d; i < TCHUNK * 38; i += DINNER) {
    int t = i / 38, k = i % 38;
    async_ld_lds_b32(&sh[t][k],
                     &xdbl[(size_t)(s * LSEQ + t0 + t) * XDBL_LD + k]);
  }
  wait_asynccnt0();
  __syncthreads();

  float Av[DSTATE], h[DSTATE];
  size_t base = (((size_t)s * NCHUNK + cch) * DINNER + d) * DSTATE;
#pragma unroll
  for (int j = 0; j < DSTATE; ++j) {
    Av[j] = -__expf(Alog[d * DSTATE + j]);
    h[j]  = Hin[base + j];
  }
  float wdt[DTRANK];
#pragma unroll
  for (int r = 0; r < DTRANK; ++r) wdt[r] = dtw[d * DTRANK + r];
  float bdt = dtb[d];
  float Dd  = Dp[d];

  for (int tt = 0; tt < TCHUNK; ++tt) {
    int t = t0 + tt;
    size_t row = (size_t)s * LSEQ + t;
    float u = xc[row * DINNER + d];
    float dtv = bdt;
#pragma unroll
    for (int r = 0; r < DTRANK; ++r) dtv = fmaf(wdt[r], sh[tt][r], dtv);
    float delta = (dtv > 20.f) ? dtv : log1pf(__expf(dtv));
    float du = delta * u;
    float y = 0.f;
#pragma unroll
    for (int j = 0; j < DSTATE; ++j) {
      float aj = __expf(delta * Av[j]);
      h[j] = fmaf(aj, h[j], du * sh[tt][6 + j]);
      y = fmaf(h[j], sh[tt][22 + j], y);
    }
    y = fmaf(Dd, u, y);
    size_t zrow = (dir < 2) ? row
                            : (size_t)((dir - 2) * 4 + bb) * LSEQ + (4095 - t);
    float z = xz01[zrow * 384 + DINNER + d];
    y *= z / (1.f + __expf(-z)); // * silu(z)
    xc[row * DINNER + d] = y;
  }
}

// ---------------------------------------------------------------------------
// K7: gated 4-direction combine -> out (B, C, H, W)
// ---------------------------------------------------------------------------
__global__ void combine_kernel(const float* __restrict__ o, const float* __restrict__ gate,
                               float* __restrict__ out) {
  int gid = blockIdx.x * blockDim.x + threadIdx.x;
  if (gid >= NPIX * CDIM) return;
  int c   = gid % CDIM;
  int pix = gid / CDIM;
  int bb  = pix >> 12;
  int rem = pix & 4095;
  int hh  = rem >> 6;
  int ww  = rem & 63;
  int l0 = hh * 64 + ww;
  int l1 = ww * 64 + hh;
  size_t r0 = (size_t)(0 * 4 + bb) * LSEQ + l0;
  size_t r1 = (size_t)(1 * 4 + bb) * LSEQ + l1;
  size_t r2 = (size_t)(2 * 4 + bb) * LSEQ + (4095 - l0);
  size_t r3 = (size_t)(3 * 4 + bb) * LSEQ + (4095 - l1);
  const float* g = gate + (size_t)pix * 4;
  float v = g[0] * o[r0 * CDIM + c] + g[1] * o[r1 * CDIM + c] +
            g[2] * o[r2 * CDIM + c] + g[3] * o[r3 * CDIM + c];
  out[((size_t)(bb * CDIM + c) * 64 + hh) * 64 + ww] = v;
}

// ---------------------------------------------------------------------------
extern "C" void kernel_launch(void* const* d_in, const int* in_sizes, int n_in,
                              void* d_out, int out_size, void* d_ws, size_t ws_size,
                              hipStream_t stream) {
  const float* x          = (const float*)d_in[0];
  const float* norm_g     = (const float*)d_in[1];
  const float* norm_b     = (const float*)d_in[2];
  const float* gate_w1    = (const float*)d_in[3];
  const float* gate_b1    = (const float*)d_in[4];
  const float* gate_w2    = (const float*)d_in[5];
  const float* gate_b2    = (const float*)d_in[6];
  const float* in_proj_w  = (const float*)d_in[7];
  const float* conv_w     = (const float*)d_in[8];
  const float* conv_b     = (const float*)d_in[9];
  const float* x_proj_w   = (const float*)d_in[10];
  const float* dt_proj_w  = (const float*)d_in[11];
  const float* dt_proj_b  = (const float*)d_in[12];
  const float* A_log      = (const float*)d_in[13];
  const float* Dp         = (const float*)d_in[14];
  const float* out_proj_w = (const float*)d_in[15];
  float* out = (float*)d_out;

  // workspace layout (floats); total = 42,532,864 floats (~162.3 MiB)
  float* ws   = (float*)d_ws;
  float* xn   = ws;                   // 16384*96       = 1,572,864
  float* gate = xn + 1572864;         // 16384*4        = 65,536
  float* xz01 = gate + 65536;         // 32768*384      = 12,582,912
  float* xc   = xz01 + 12582912;      // 65536*192      = 12,582,912 (reused as y_gated)
  float* xdbl = xc + 12582912;        // 65536*48       = 3,145,728
  float* o    = xdbl + 3145728;       // 65536*96       = 6,291,456
  float* P    = o + 6291456;          // 16*64*192*16   = 3,145,728
  float* Hloc = P + 3145728;          // 16*64*192*16   = 3,145,728

  // 1) LayerNorm + gate
  ln_gate_kernel<<<NPIX, CDIM, 0, stream>>>(x, norm_g, norm_b, gate_w1, gate_b1,
                                            gate_w2, gate_b2, xn, gate);
  // 2) in_proj (dirs 0,1): M=32768, N=384, K=96
  inproj_wmma<<<dim3(32768 / 16, 384 / 16), 32, 0, stream>>>(xn, in_proj_w, xz01);
  // 3) causal depthwise conv + SiLU (all 16 sequences)
  conv_silu_kernel<<<(NSEQ * LSEQ * DINNER) / 256, 256, 0, stream>>>(xz01, conv_w,
                                                                     conv_b, xc);
  // 4) x_proj: M=65536, K=192, N=38 (padded to 48)
  gemm_wmma_nt<<<dim3(65536 / 16, XDBL_LD / 16), 32, 0, stream>>>(
      xc, DINNER, x_proj_w, DINNER, 38, xdbl, XDBL_LD, DINNER);
  // 5) chunked selective scan (3 phases)
  scan_pass1<<<dim3(NSEQ, NCHUNK), DINNER, 0, stream>>>(xc, xdbl, dt_proj_w,
                                                        dt_proj_b, A_log, P, Hloc);
  scan_combine<<<(NSEQ * DINNER * DSTATE + 255) / 256, 256, 0, stream>>>(P, Hloc);
  scan_pass2<<<dim3(NSEQ, NCHUNK), DINNER, 0, stream>>>(xc, xdbl, xz01, dt_proj_w,
                                                        dt_proj_b, A_log, Dp, Hloc);
  // 6) out_proj: M=65536, K=192, N=96
  gemm_wmma_nt<<<dim3(65536 / 16, CDIM / 16), 32, 0, stream>>>(
      xc, DINNER, out_proj_w, DINNER, CDIM, o, CDIM, DINNER);
  // 7) gated directional combine
  combine_kernel<<<(NPIX * CDIM + 255) / 256, 256, 0, stream>>>(o, gate, out);
}